// HypergraphNet_50895362458091
// MI455X (gfx1250) — compile-verified
//
#include <hip/hip_runtime.h>

#define FEAT 128

typedef __attribute__((ext_vector_type(2))) float v2f;
typedef __attribute__((ext_vector_type(8))) float v8f;

// ---------------------------------------------------------------- utilities

__global__ __launch_bounds__(256) void zero_f32v4(float4* __restrict__ p, long n4) {
  long i = (long)blockIdx.x * blockDim.x + threadIdx.x;
  long stride = (long)gridDim.x * blockDim.x;
  for (; i < n4; i += stride) p[i] = make_float4(0.f, 0.f, 0.f, 0.f);
}

// deg_n[n] += w[e]; deg_e[e] += 1   (segment sums for D and B)
__global__ __launch_bounds__(256) void degrees_kernel(
    const int* __restrict__ node_idx, const int* __restrict__ edge_idx,
    const float* __restrict__ w, float* __restrict__ deg_n,
    float* __restrict__ deg_e, int nnz) {
  int i = blockIdx.x * blockDim.x + threadIdx.x;
  if (i >= nnz) return;
  int n = node_idx[i];
  int e = edge_idx[i];
  atomicAdd(&deg_n[n], w[e]);
  atomicAdd(&deg_e[e], 1.0f);
}

// d[i] = d[i] > 0 ? 1/d[i] : 0   (D^-1, in place)
__global__ __launch_bounds__(256) void invert_deg_kernel(float* __restrict__ d, int n) {
  int i = blockIdx.x * blockDim.x + threadIdx.x;
  if (i < n) { float v = d[i]; d[i] = (v > 0.f) ? (1.f / v) : 0.f; }
}

// c[e] = deg_e[e] > 0 ? w[e]/deg_e[e] : 0   (fold w_e * B^-1 into one scalar)
__global__ __launch_bounds__(256) void edge_coef_kernel(
    float* __restrict__ c, const float* __restrict__ w, int n) {
  int i = blockIdx.x * blockDim.x + threadIdx.x;
  if (i < n) { float v = c[i]; c[i] = (v > 0.f) ? (w[i] / v) : 0.f; }
}

// ---------------------------------------------------------------- WMMA GEMM
// h[M,128] = x[M,128] @ W[128,128] using V_WMMA_F32_16X16X4_F32.
// One wave computes a 16-row x 128-col strip (8 accumulator tiles),
// K-loop in steps of 4.  f32 A 16x4 layout: lane<16 -> (M=lane, K=0/1),
// lane>=16 -> (M=lane-16, K=2/3). B 4x16 mirrored (N=lane%16).
__global__ __launch_bounds__(256) void gemm_xw_wmma(
    const float* __restrict__ x, const float* __restrict__ W,
    float* __restrict__ h, int M) {
  int gid  = blockIdx.x * blockDim.x + threadIdx.x;
  int wave = gid >> 5;
  int lane = gid & 31;
  int row0 = wave << 4;
  if (row0 >= M) return;            // uniform per wave: EXEC all-ones at WMMA
  int lm = lane & 15;               // M index in A tile / N index in B tile
  int kg = (lane >> 4) << 1;        // K sub-offset: 0 (lanes 0-15) or 2 (16-31)

  v8f acc[8] = {};
  const float* xrow = x + (size_t)(row0 + lm) * FEAT;

  #pragma unroll 4
  for (int k0 = 0; k0 < FEAT; k0 += 4) {
    v2f a = *(const v2f*)(xrow + k0 + kg);            // contiguous float2
    const float* wr = W + (size_t)(k0 + kg) * FEAT + lm;
    #pragma unroll
    for (int t = 0; t < 8; ++t) {
      v2f b;
      b.x = wr[t * 16];
      b.y = wr[FEAT + t * 16];
      acc[t] = __builtin_amdgcn_wmma_f32_16x16x4_f32(
          false, a, false, b, (short)0, acc[t], false, false);
    }
  }

  // C/D layout: VGPR r: lanes 0-15 -> (M=r, N=lane); lanes 16-31 -> (M=8+r).
  int mbase = (lane >> 4) << 3;     // 0 or 8
  float* hbase = h + (size_t)row0 * FEAT + lm;
  #pragma unroll
  for (int t = 0; t < 8; ++t) {
    #pragma unroll
    for (int r = 0; r < 8; ++r) {
      hbase[(size_t)(mbase + r) * FEAT + t * 16] = acc[t][r];
    }
  }
}

// ---------------------------------------------------------------- scatters
// One wave per nnz entry; lane handles a float4 chunk of the 128-float row.
// Gather is a fully-coalesced 512B contiguous read; scatter is 512B of
// contiguous global_atomic_add_f32.

__global__ __launch_bounds__(256) void scatter_node_to_edge(
    const float* __restrict__ h, const int* __restrict__ node_idx,
    const int* __restrict__ edge_idx, float* __restrict__ eacc, int nnz) {
  int gid = blockIdx.x * blockDim.x + threadIdx.x;
  int i = gid >> 5;
  if (i >= nnz) return;
  int lane4 = (gid & 31) << 2;
  int n = node_idx[i];
  int e = edge_idx[i];
  float4 v = *(const float4*)(h + (size_t)n * FEAT + lane4);
  float* dst = eacc + (size_t)e * FEAT + lane4;
  atomicAdd(dst + 0, v.x);
  atomicAdd(dst + 1, v.y);
  atomicAdd(dst + 2, v.z);
  atomicAdd(dst + 3, v.w);
}

__global__ __launch_bounds__(256) void scatter_edge_to_node(
    const float* __restrict__ eacc, const float* __restrict__ coef,
    const int* __restrict__ node_idx, const int* __restrict__ edge_idx,
    float* __restrict__ nacc, int nnz) {
  int gid = blockIdx.x * blockDim.x + threadIdx.x;
  int i = gid >> 5;
  if (i >= nnz) return;
  int lane4 = (gid & 31) << 2;
  int n = node_idx[i];
  int e = edge_idx[i];
  float c = coef[e];
  float4 v = *(const float4*)(eacc + (size_t)e * FEAT + lane4);
  float* dst = nacc + (size_t)n * FEAT + lane4;
  atomicAdd(dst + 0, v.x * c);
  atomicAdd(dst + 1, v.y * c);
  atomicAdd(dst + 2, v.z * c);
  atomicAdd(dst + 3, v.w * c);
}

// y = y * d_inv[node] + bias[col], optional ReLU (in place)
__global__ __launch_bounds__(256) void finalize_kernel(
    float* __restrict__ y, const float* __restrict__ dinv,
    const float* __restrict__ bias, int n_nodes, int do_relu) {
  int gid = blockIdx.x * blockDim.x + threadIdx.x;
  if (gid >= n_nodes * (FEAT / 4)) return;
  int node = gid >> 5;
  int f4 = (gid & 31) << 2;
  float s = dinv[node];
  float4* yp = (float4*)(y + (size_t)node * FEAT + f4);
  float4 v = *yp;
  float4 b = *(const float4*)(bias + f4);
  v.x = fmaf(v.x, s, b.x);
  v.y = fmaf(v.y, s, b.y);
  v.z = fmaf(v.z, s, b.z);
  v.w = fmaf(v.w, s, b.w);
  if (do_relu) {
    v.x = fmaxf(v.x, 0.f);
    v.y = fmaxf(v.y, 0.f);
    v.z = fmaxf(v.z, 0.f);
    v.w = fmaxf(v.w, 0.f);
  }
  *yp = v;
}

// ---------------------------------------------------------------- launch

extern "C" void kernel_launch(void* const* d_in, const int* in_sizes, int n_in,
                              void* d_out, int out_size, void* d_ws, size_t ws_size,
                              hipStream_t stream) {
  const float* x    = (const float*)d_in[0];
  const int*   hidx = (const int*)d_in[1];     // [2, NNZ]
  const float* wgt  = (const float*)d_in[2];   // [E]
  const float* W1   = (const float*)d_in[3];
  const float* b1   = (const float*)d_in[4];
  const float* W2   = (const float*)d_in[5];
  const float* b2   = (const float*)d_in[6];
  float* out = (float*)d_out;

  const int Nn  = in_sizes[0] / FEAT;          // 65536 nodes
  const int Ne  = in_sizes[2];                 // 65536 hyperedges
  const int nnz = in_sizes[1] / 2;             // 2^20

  const int* node_idx = hidx;
  const int* edge_idx = hidx + nnz;

  // workspace layout (floats)
  float* hbuf = (float*)d_ws;                  // [Nn,128] GEMM output
  float* ebuf = hbuf + (size_t)Nn * FEAT;      // [Ne,128] edge accumulator
  float* ybuf = ebuf + (size_t)Ne * FEAT;      // [Nn,128] layer-1 output
  float* dinv = ybuf + (size_t)Nn * FEAT;      // [Nn] 1/deg_n
  float* coef = dinv + Nn;                     // [Ne] w_e/deg_e

  const int T = 256;
  dim3 blk(T);
  long nfeat4_n = (long)Nn * (FEAT / 4);
  long nfeat4_e = (long)Ne * (FEAT / 4);
  dim3 gZeroN((unsigned)((nfeat4_n + T - 1) / T));
  dim3 gZeroE((unsigned)((nfeat4_e + T - 1) / T));
  dim3 gZeroDeg((unsigned)(((Nn + Ne) / 4 + T - 1) / T));
  dim3 gNnz((unsigned)((nnz + T - 1) / T));
  dim3 gNnzWave((unsigned)(((size_t)nnz * 32 + T - 1) / T));
  dim3 gGemm((unsigned)(((size_t)(Nn / 16) * 32 + T - 1) / T));
  dim3 gFin((unsigned)((Nn * (FEAT / 4) + T - 1) / T));
  dim3 gN((unsigned)((Nn + T - 1) / T));
  dim3 gE((unsigned)((Ne + T - 1) / T));

  // --- degree terms (shared by both layers): dinv/coef used as accumulators
  zero_f32v4<<<gZeroDeg, blk, 0, stream>>>((float4*)dinv, (long)(Nn + Ne) / 4);
  degrees_kernel<<<gNnz, blk, 0, stream>>>(node_idx, edge_idx, wgt, dinv, coef, nnz);
  invert_deg_kernel<<<gN, blk, 0, stream>>>(dinv, Nn);
  edge_coef_kernel<<<gE, blk, 0, stream>>>(coef, wgt, Ne);

  // --- layer 1 ---
  gemm_xw_wmma<<<gGemm, blk, 0, stream>>>(x, W1, hbuf, Nn);
  zero_f32v4<<<gZeroE, blk, 0, stream>>>((float4*)ebuf, nfeat4_e);
  scatter_node_to_edge<<<gNnzWave, blk, 0, stream>>>(hbuf, node_idx, edge_idx, ebuf, nnz);
  zero_f32v4<<<gZeroN, blk, 0, stream>>>((float4*)ybuf, nfeat4_n);
  scatter_edge_to_node<<<gNnzWave, blk, 0, stream>>>(ebuf, coef, node_idx, edge_idx, ybuf, nnz);
  finalize_kernel<<<gFin, blk, 0, stream>>>(ybuf, dinv, b1, Nn, /*relu=*/1);

  // --- layer 2 ---
  gemm_xw_wmma<<<gGemm, blk, 0, stream>>>(ybuf, W2, hbuf, Nn);
  zero_f32v4<<<gZeroE, blk, 0, stream>>>((float4*)ebuf, nfeat4_e);
  scatter_node_to_edge<<<gNnzWave, blk, 0, stream>>>(hbuf, node_idx, edge_idx, ebuf, nnz);
  zero_f32v4<<<gZeroN, blk, 0, stream>>>((float4*)out, nfeat4_n);
  scatter_edge_to_node<<<gNnzWave, blk, 0, stream>>>(ebuf, coef, node_idx, edge_idx, out, nnz);
  finalize_kernel<<<gFin, blk, 0, stream>>>(out, dinv, b2, Nn, /*relu=*/0);
}